// GramSchmidt_19980187861913
// MI455X (gfx1250) — compile-verified
//
#include <hip/hip_runtime.h>

typedef float v2f __attribute__((ext_vector_type(2)));
typedef float v4f __attribute__((ext_vector_type(4)));
typedef float v8f __attribute__((ext_vector_type(8)));

#define GS_B  512
#define GS_K  64
#define GS_L  1024
#define GS_LP 1028          // padded LDS row stride (floats) to break bank alignment
#define GS_NT 1024          // 32 waves

// Intra-wave LDS producer->consumer fence (cross-lane through LDS).
__device__ __forceinline__ void lds_fence() {
  asm volatile("s_wait_dscnt 0" ::: "memory");
  __builtin_amdgcn_wave_barrier();
}

__global__ __launch_bounds__(GS_NT, 1)
void gram_schmidt_cholqr(const float* __restrict__ xin,
                         float* __restrict__ xout) {
  extern __shared__ float smem[];
  float* X = smem;                      // 64 x 1028 fp32  (257 KB)
  float* G = smem + GS_K * GS_LP;       // 64 x 64 Gram -> Cholesky factor L
  float* W = G + GS_K * GS_K;           // 64 x 64 partials -> L^{-1}

  const int b    = blockIdx.x;
  const int tid  = threadIdx.x;
  const int wave = tid >> 5;            // wave32
  const int lane = tid & 31;

  const float* xg = xin  + (size_t)b * (GS_K * GS_L);
  float*       og = xout + (size_t)b * (GS_K * GS_L);

  // ---------- Phase 1: global -> LDS, 128-bit transfers, padded rows ----------
  for (int idx = tid; idx < GS_K * (GS_L / 4); idx += GS_NT) {
    int r  = idx >> 8;                  // row 0..63
    int c4 = (idx & 255) << 2;          // col (float4 granules)
    *(v4f*)(X + r * GS_LP + c4) = *(const v4f*)(xg + r * GS_L + c4);
  }
  __syncthreads();

  // ---------- Phase 2: G = X * X^T via v_wmma_f32_16x16x4_f32 ----------
  // 16 output tiles (4x4 grid of 16x16), 2 waves per tile splitting K=1024.
  {
    const int tile  = wave >> 1;
    const int ti    = tile >> 2;
    const int tj    = tile & 3;
    const int khalf = wave & 1;
    const int m     = lane & 15;
    const int koff  = (lane < 16) ? 0 : 2;   // A/B f32 layout: VGPR0=K{0|2}, VGPR1=K{1|3}
    const float* pa = X + (ti * 16 + m) * GS_LP + koff + khalf * (GS_L / 2);
    const float* pb = X + (tj * 16 + m) * GS_LP + koff + khalf * (GS_L / 2);
    v8f acc = {0.f, 0.f, 0.f, 0.f, 0.f, 0.f, 0.f, 0.f};
    #pragma unroll 4
    for (int k = 0; k < GS_L / 2; k += 4) {
      v2f a  = *(const v2f*)(pa + k);        // ds_load_b64
      v2f bb = *(const v2f*)(pb + k);        // ds_load_b64
      acc = __builtin_amdgcn_wmma_f32_16x16x4_f32(false, a, false, bb,
                                                  (short)0, acc, false, false);
    }
    // C/D layout: VGPR j -> (M = j | j+8, N = lane&15)
    float*    dst = khalf ? W : G;           // khalf=1 partials land in W temp
    const int mb  = (lane < 16) ? 0 : 8;
    #pragma unroll
    for (int j = 0; j < 8; ++j)
      dst[(ti * 16 + mb + j) * GS_K + tj * 16 + m] = acc[j];
  }
  __syncthreads();
  for (int idx = tid; idx < GS_K * GS_K; idx += GS_NT) G[idx] += W[idx];
  __syncthreads();

  // ---------- Phase 3: in-place Cholesky of G (lower triangle), wave 0 ----------
  if (wave == 0) {
    for (int j = 0; j < GS_K; ++j) {
      float ljj = sqrtf(G[j * GS_K + j]);
      float inv = 1.0f / ljj;
      if (lane == 0) G[j * GS_K + j] = ljj;
      for (int i = j + 1 + lane; i < GS_K; i += 32)
        G[i * GS_K + j] *= inv;
      lds_fence();
      for (int i = j + 1 + lane; i < GS_K; i += 32) {
        float lij = G[i * GS_K + j];
        for (int k = j + 1; k <= i; ++k)
          G[i * GS_K + k] -= lij * G[k * GS_K + j];
      }
      lds_fence();
    }
  }
  __syncthreads();

  // ---------- Phase 4: W = L^{-1}, one column per (wave, pass) ----------
  for (int j = wave; j < GS_K; j += 32) {
    for (int i = lane; i < GS_K; i += 32)
      W[i * GS_K + j] = (i == j) ? 1.0f : 0.0f;
    lds_fence();
    for (int k = j; k < GS_K; ++k) {
      float wk = W[k * GS_K + j] / G[k * GS_K + k];   // all lanes redundantly
      if (lane == 0) W[k * GS_K + j] = wk;
      for (int i = k + 1 + lane; i < GS_K; i += 32)
        W[i * GS_K + j] -= G[i * GS_K + k] * wk;
      lds_fence();
    }
  }
  __syncthreads();

  // ---------- Phase 5: Q = W * X -> global (rows already unit-norm) ----------
  {
    const int m    = lane & 15;
    const int koff = (lane < 16) ? 0 : 2;
    const int mb   = (lane < 16) ? 0 : 8;
    for (int t = wave; t < 4 * (GS_L / 16); t += 32) {   // 256 tiles, 8 per wave
      const int ti = t >> 6;                             // row tile 0..3
      const int tc = t & 63;                             // col tile 0..63
      const float* pa = W + (ti * 16 + m) * GS_K + koff;
      v8f acc = {0.f, 0.f, 0.f, 0.f, 0.f, 0.f, 0.f, 0.f};
      #pragma unroll
      for (int k0 = 0; k0 < GS_K; k0 += 4) {
        v2f a = *(const v2f*)(pa + k0);                  // A from W (row-major)
        v2f bb;                                          // B rows of X (K-strided)
        bb.x = X[(k0 + koff    ) * GS_LP + tc * 16 + m];
        bb.y = X[(k0 + koff + 1) * GS_LP + tc * 16 + m];
        acc = __builtin_amdgcn_wmma_f32_16x16x4_f32(false, a, false, bb,
                                                    (short)0, acc, false, false);
      }
      #pragma unroll
      for (int j = 0; j < 8; ++j)
        og[(ti * 16 + mb + j) * GS_L + tc * 16 + m] = acc[j];
    }
  }
}

extern "C" void kernel_launch(void* const* d_in, const int* in_sizes, int n_in,
                              void* d_out, int out_size, void* d_ws, size_t ws_size,
                              hipStream_t stream) {
  (void)in_sizes; (void)n_in; (void)out_size; (void)d_ws; (void)ws_size;
  const float* x   = (const float*)d_in[0];
  float*       out = (float*)d_out;

  const size_t shmem = (size_t)(GS_K * GS_LP + 2 * GS_K * GS_K) * sizeof(float); // 295,936 B < 320 KB
  hipFuncSetAttribute(reinterpret_cast<const void*>(&gram_schmidt_cholqr),
                      hipFuncAttributeMaxDynamicSharedMemorySize, (int)shmem);

  gram_schmidt_cholqr<<<dim3(GS_B), dim3(GS_NT), shmem, stream>>>(x, out);
}